// ConvLRULayer_13821204758546
// MI455X (gfx1250) — compile-verified
//
#include <hip/hip_runtime.h>
#include <math.h>

// CDNA5 fp32 WMMA operand types
typedef __attribute__((ext_vector_type(2))) float v2f;   // A/B: 16x4 or 4x16 f32 = 2 VGPRs
typedef __attribute__((ext_vector_type(8))) float v8f;   // C/D: 16x16 f32 = 8 VGPRs

#define ANG_STEP 0.049087385212340526f   // 2*pi/128

__device__ __forceinline__ v8f wmma_f32(v2f a, v2f b, v8f c) {
  // D = A(16x4) * B(4x16) + C(16x16), fp32 throughout
  return __builtin_amdgcn_wmma_f32_16x16x4_f32(
      /*neg_a=*/false, a, /*neg_b=*/false, b,
      /*c_mod=*/(short)0, c, /*reuse_a=*/false, /*reuse_b=*/false);
}

// x:        [B=2, L=32, C=32, S=64, W=128] f32
// nu_log:   [C, S, Wf=65] f32
// theta_log:[C, S, Wf=65] f32
// out:      [B, L, C, S, W] f32
// One wave (32 threads) fully processes one (b,c,s) sequence.
__launch_bounds__(32)
__global__ void conv_lru_fused(const float* __restrict__ x,
                               const float* __restrict__ nu_log,
                               const float* __restrict__ theta_log,
                               float* __restrict__ out) {
  // LDS staging: x tile, and weighted scan state (forward results overwritten in place)
  __shared__ float xs[32][132];   // 32 rows of 128, stride 132 -> conflict-free A fetch
  __shared__ float hre[32][68];   // P then wk*Re(h)
  __shared__ float him[32][68];   // Q then -wk*Im(h)
  __shared__ float h64[32];       // Nyquist: P64 then (1/128)*Re(h64)

  const int t   = threadIdx.x;          // 0..31
  const int seq = blockIdx.x;           // 0..4095 = b*2048 + c*64 + s
  const int b   = seq >> 11;
  const int cs  = seq & 2047;           // c*64 + s

  // ---- phase 0: stage x (coalesced 512B rows) ----
  for (int l = 0; l < 32; ++l) {
    const float4 v = *(const float4*)(x + ((size_t)((b * 32 + l) * 2048 + cs)) * 128 + 4 * t);
    *(float4*)&xs[l][4 * t] = v;
  }
  __syncthreads();

  // ---- phase 0b: Nyquist projection P64[l] = sum_w x[l][w]*(-1)^w (lane t = row t) ----
  {
    float p = 0.f;
    #pragma unroll
    for (int w = 0; w < 128; w += 2) p += xs[t][w] - xs[t][w + 1];
    h64[t] = p;
  }

  const int m  = t & 15;   // WMMA row within tile
  const int kh = t >> 4;   // K half select (ISA f32 A/B layout)

  // ---- phase 1: forward DFT via WMMA: P[l,k]=sum_w x*cos, Q[l,k]=sum_w x*sin ----
  for (int nt = 0; nt < 4; ++nt) {                 // mode tiles k = 0..63
    v8f P0 = {0,0,0,0,0,0,0,0};
    v8f P1 = P0, Q0 = P0, Q1 = P0;
    const int kmode = nt * 16 + m;                 // this lane's N column (mode)
    for (int ks = 0; ks < 32; ++ks) {              // K = w in chunks of 4
      const int w0 = 4 * ks + 2 * kh;
      const v2f a0 = *(const v2f*)&xs[m][w0];       // A rows l=0..15
      const v2f a1 = *(const v2f*)&xs[16 + m][w0];  // A rows l=16..31
      const float ang0 = (float)((kmode * w0) & 127) * ANG_STEP;
      const float ang1 = (float)((kmode * (w0 + 1)) & 127) * ANG_STEP;
      const v2f bc = { __cosf(ang0), __cosf(ang1) };
      const v2f bs = { __sinf(ang0), __sinf(ang1) };
      P0 = wmma_f32(a0, bc, P0);
      Q0 = wmma_f32(a0, bs, Q0);
      P1 = wmma_f32(a1, bc, P1);
      Q1 = wmma_f32(a1, bs, Q1);
    }
    // C layout: vgpr r, lane t -> row r+8*kh, col (t&15)
    const int col = nt * 16 + m;
    #pragma unroll
    for (int r = 0; r < 8; ++r) {
      const int lo = r + 8 * kh;
      hre[lo][col]      = P0[r];
      hre[16 + lo][col] = P1[r];
      him[lo][col]      = Q0[r];
      him[16 + lo][col] = Q1[r];
    }
  }
  __syncthreads();

  // ---- phase 2: diagonal LRU scan per mode, irfft weights folded in ----
  // lam = exp(-exp(nu)) * exp(i*exp(theta)); bx = gamma*(P - iQ)
  const float* nu = nu_log    + (size_t)cs * 65;
  const float* th = theta_log + (size_t)cs * 65;
  for (int pass = 0; pass < 2; ++pass) {
    const int k   = t + 32 * pass;                 // modes 0..63
    const float aa = __expf(-__expf(nu[k]));
    const float te = __expf(th[k]);
    const float lr = aa * __cosf(te);
    const float li = aa * __sinf(te);
    const float g  = sqrtf(fmaxf(1.f - aa * aa, 0.f));
    const float wk = (k == 0) ? (1.f / 128.f) : (2.f / 128.f);
    float hr = 0.f, hi = 0.f;
    for (int l = 0; l < 32; ++l) {
      const float p = hre[l][k];
      const float q = him[l][k];
      const float nhr = lr * hr - li * hi + g * p;
      const float nhi = li * hr + lr * hi - g * q;
      hr = nhr; hi = nhi;
      hre[l][k] =  wk * hr;    // wk*Re(h)
      him[l][k] = -wk * hi;    // -wk*Im(h) so inverse is +A_im*sin
    }
  }
  if (t == 0) {                                    // Nyquist mode k=64 (X is real)
    const float aa = __expf(-__expf(nu[64]));
    const float te = __expf(th[64]);
    const float lr = aa * __cosf(te);
    const float li = aa * __sinf(te);
    const float g  = sqrtf(fmaxf(1.f - aa * aa, 0.f));
    float hr = 0.f, hi = 0.f;
    for (int l = 0; l < 32; ++l) {
      const float p = h64[l];
      const float nhr = lr * hr - li * hi + g * p;
      const float nhi = li * hr + lr * hi;
      hr = nhr; hi = nhi;
      h64[l] = hr * (1.f / 128.f);                 // sin(pi*w)=0 kills Im part
    }
  }
  __syncthreads();

  // ---- phase 3: inverse DFT via WMMA: out[l,w] = sum_k (wk*Hre)*cos + (-wk*Him)*sin ----
  for (int nt = 0; nt < 8; ++nt) {                 // w tiles 0..127
    v8f o0 = {0,0,0,0,0,0,0,0};
    v8f o1 = o0;
    const int w = nt * 16 + m;                     // this lane's N column (w)
    for (int ks = 0; ks < 16; ++ks) {              // K = mode in chunks of 4
      const int k0 = 4 * ks + 2 * kh;
      const v2f ar0 = *(const v2f*)&hre[m][k0];
      const v2f ar1 = *(const v2f*)&hre[16 + m][k0];
      const v2f ai0 = *(const v2f*)&him[m][k0];
      const v2f ai1 = *(const v2f*)&him[16 + m][k0];
      const float ang0 = (float)((k0 * w) & 127) * ANG_STEP;
      const float ang1 = (float)(((k0 + 1) * w) & 127) * ANG_STEP;
      const v2f bc = { __cosf(ang0), __cosf(ang1) };
      const v2f bs = { __sinf(ang0), __sinf(ang1) };
      o0 = wmma_f32(ar0, bc, o0);
      o0 = wmma_f32(ai0, bs, o0);
      o1 = wmma_f32(ar1, bc, o1);
      o1 = wmma_f32(ai1, bs, o1);
    }
    const float sg = (w & 1) ? -1.f : 1.f;         // Nyquist cos(pi*w)
    #pragma unroll
    for (int r = 0; r < 8; ++r) {
      const int l0 = r + 8 * kh;
      const int l1 = 16 + l0;
      out[((size_t)((b * 32 + l0) * 2048 + cs)) * 128 + w] = o0[r] + sg * h64[l0];
      out[((size_t)((b * 32 + l1) * 2048 + cs)) * 128 + w] = o1[r] + sg * h64[l1];
    }
  }
}

extern "C" void kernel_launch(void* const* d_in, const int* in_sizes, int n_in,
                              void* d_out, int out_size, void* d_ws, size_t ws_size,
                              hipStream_t stream) {
  (void)in_sizes; (void)n_in; (void)d_ws; (void)ws_size; (void)out_size;
  const float* x  = (const float*)d_in[0];
  const float* nu = (const float*)d_in[1];
  const float* th = (const float*)d_in[2];
  float* out = (float*)d_out;
  dim3 grid(4096);   // one wave per (b,c,s) sequence
  dim3 block(32);    // wave32
  hipLaunchKernelGGL(conv_lru_fused, grid, block, 0, stream, x, nu, th, out);
}